// DeltaNet_41961830482331
// MI455X (gfx1250) — compile-verified
//
#include <hip/hip_runtime.h>

typedef float v2f __attribute__((ext_vector_type(2)));
typedef float v8f __attribute__((ext_vector_type(8)));

#define NH    12
#define HS    64
#define TLEN  1024
#define BSZ   4
#define CDIM  768

#define LDSB_STRIDE 68   // 16 rows x 68 floats; 68*4B = 272B = 17*16B (b128-aligned rows)

// ---------------------------------------------------------------------------
// FP32 WMMA GEMM: D[M][N] = A[M][K] @ B[K][N] (+ optional residual)
// Block = 128 threads (4 waves) computing a 64x64 tile; waves stacked on M
// (16 rows each), all four share the B panel staged in LDS (double-buffered,
// K-chunk = 16 -> 4 WMMA K-steps per stage).
//
// fp32 A 16x4 fragment (2 VGPRs): lane<16 holds A[row=lane][k+0/(+1)],
// lane>=16 holds A[row=lane-16][k+2/(+3)]  -> one float2 global load per lane.
// fp32 B 4x16 fragment (2 VGPRs): v0 = B[k+2*(lane/16)][n0+lane%16],
// v1 = B[k+1+2*(lane/16)][n0+lane%16]      -> two ds loads from staged panel.
// ---------------------------------------------------------------------------
__global__ void __launch_bounds__(128)
wmma_gemm_f32(const float* __restrict__ A, const float* __restrict__ Bm,
              const float* __restrict__ Resid, float* __restrict__ D,
              int M, int N, int K)
{
    __shared__ __align__(16) float ldsB[2][16 * LDSB_STRIDE];

    const int tid  = threadIdx.x;
    const int lane = tid & 31;
    const int wv   = tid >> 5;                  // wave within block: M sub-tile
    const int ntiles = N >> 6;
    const int tm = (int)blockIdx.x / ntiles;
    const int tn = (int)blockIdx.x - tm * ntiles;
    const int m0 = tm * 64 + wv * 16;
    const int n0 = tn * 64;
    if (m0 >= M) return;

    const int ar = lane & 15;                   // A row within 16-row tile
    const int ak = (lane >> 4) << 1;            // 0 or 2: K sub-offset
    const int bn = lane & 15;                   // B col within 16-col group

    // cooperative B staging coords: thread -> rows (sr, sr+8), 4 cols at sc
    const int sr = tid >> 4;                    // 0..7
    const int sc = (tid & 15) << 2;             // 0,4,...,60

    v8f acc[4];
    #pragma unroll
    for (int j = 0; j < 4; ++j) acc[j] = (v8f){};

    const float* arow   = A  + (size_t)(m0 + ar) * K + ak;
    const float* bstage = Bm + (size_t)sr * N + n0 + sc;

    // prologue: stage K-chunk 0 into buffer 0
    {
        float4 x0 = *(const float4*)(bstage);
        float4 x1 = *(const float4*)(bstage + (size_t)8 * N);
        *(float4*)&ldsB[0][sr * LDSB_STRIDE + sc]       = x0;
        *(float4*)&ldsB[0][(sr + 8) * LDSB_STRIDE + sc] = x1;
    }
    __syncthreads();

    const int nstages = K >> 4;                 // K-chunks of 16
    for (int s = 0; s < nstages; ++s) {
        const int cur = s & 1;
        const int nxt = cur ^ 1;
        const int k0  = s << 4;

        if (s + 2 < nstages)                    // gfx1250 global_prefetch_b8
            __builtin_prefetch(bstage + (size_t)(k0 + 32) * N, 0, 1);

        if (s + 1 < nstages) {                  // stage next chunk
            const float* bs = bstage + (size_t)(k0 + 16) * N;
            float4 x0 = *(const float4*)(bs);
            float4 x1 = *(const float4*)(bs + (size_t)8 * N);
            *(float4*)&ldsB[nxt][sr * LDSB_STRIDE + sc]       = x0;
            *(float4*)&ldsB[nxt][(sr + 8) * LDSB_STRIDE + sc] = x1;
        }

        const float* Lb = ldsB[cur];
        #pragma unroll
        for (int kk = 0; kk < 4; ++kk) {
            v2f afrag = *(const v2f*)(arow + k0 + 4 * kk);
            const int kl = 4 * kk + ak;
            #pragma unroll
            for (int j = 0; j < 4; ++j) {
                v2f bf;
                bf.x = Lb[kl * LDSB_STRIDE       + j * 16 + bn];
                bf.y = Lb[(kl + 1) * LDSB_STRIDE + j * 16 + bn];
                acc[j] = __builtin_amdgcn_wmma_f32_16x16x4_f32(
                    false, afrag, false, bf, (short)0, acc[j], false, false);
            }
        }
        __syncthreads();
    }

    // D 16x16 layout: VGPR i, lane l -> row = i + 8*(l/16), col = l%16
    const int row0 = m0 + ((lane >> 4) << 3);
    #pragma unroll
    for (int j = 0; j < 4; ++j) {
        const int col = n0 + j * 16 + bn;
        #pragma unroll
        for (int i = 0; i < 8; ++i) {
            const size_t idx = (size_t)(row0 + i) * N + col;
            float val = acc[j][i];
            if (Resid) val += Resid[idx];
            D[idx] = val;
        }
    }
}

// ---------------------------------------------------------------------------
// Per-64-element standardization in place (mean / unbiased std, ddof=1).
// One wave per 64-float contiguous segment; 2 elements per lane.
// ---------------------------------------------------------------------------
__global__ void __launch_bounds__(256)
qkv_norm(float* __restrict__ Q, int nseg)
{
    const int wave = (int)((blockIdx.x * blockDim.x + threadIdx.x) >> 5);
    const int lane = threadIdx.x & 31;
    if (wave >= nseg) return;
    float* p = Q + (size_t)wave * 64 + lane * 2;
    v2f v = *(v2f*)p;
    float s  = v.x + v.y;
    float sq = v.x * v.x + v.y * v.y;
    #pragma unroll
    for (int off = 16; off > 0; off >>= 1) {
        s  += __shfl_xor(s,  off);
        sq += __shfl_xor(sq, off);
    }
    const float mean = s * (1.0f / 64.0f);
    const float var  = (sq - 64.0f * mean * mean) * (1.0f / 63.0f);
    const float rstd = rsqrtf(var);
    v.x = (v.x - mean) * rstd;
    v.y = (v.y - mean) * rstd;
    *(v2f*)p = v;
}

// ---------------------------------------------------------------------------
// DeltaNet scan. One block per (b,h). S (64x64 f32) lives in registers:
// thread (r = tid/4, c = tid%4) owns S[r][16c..16c+15].
// Per step: Sk = S k (shfl_xor reduce over the 4 row lanes),
//           S  = 0.99 S - 0.01 (Sk) k^T + k v^T,
//           o  = S q (reduce), lane c==0 writes o[r].
// k,v,q staged through LDS each step.
// ---------------------------------------------------------------------------
__global__ void __launch_bounds__(256)
deltanet_scan(const float* __restrict__ qkv, float* __restrict__ yraw)
{
    __shared__ __align__(16) float sh[192];   // [k:64][v:64][q:64]
    const int bh = blockIdx.x;                // 0..47
    const int b  = bh / NH;
    const int h  = bh - b * NH;
    const int tid = threadIdx.x;
    const int r = tid >> 2;
    const int c = tid & 3;
    const int colBase = c * 16;

    float S[16];
    #pragma unroll
    for (int i = 0; i < 16; ++i) S[i] = 0.0f;

    for (int t = 0; t < TLEN; ++t) {
        const size_t base = (size_t)(b * TLEN + t) * (3 * CDIM) + h * HS;
        if (tid < 64)        sh[tid] = qkv[base + CDIM     + tid];          // k
        else if (tid < 128)  sh[tid] = qkv[base + 2 * CDIM + (tid - 64)];   // v
        else if (tid < 192)  sh[tid] = qkv[base             + (tid - 128)]; // q
        __syncthreads();

        const float kr = sh[r];
        float kc[16], vc[16], qc[16];
        #pragma unroll
        for (int i = 0; i < 16; i += 4) {
            *(float4*)(kc + i) = *(const float4*)(sh       + colBase + i);
            *(float4*)(vc + i) = *(const float4*)(sh + 64  + colBase + i);
            *(float4*)(qc + i) = *(const float4*)(sh + 128 + colBase + i);
        }

        float sk = 0.0f;
        #pragma unroll
        for (int i = 0; i < 16; ++i) sk += S[i] * kc[i];
        sk += __shfl_xor(sk, 1);
        sk += __shfl_xor(sk, 2);

        float o = 0.0f;
        #pragma unroll
        for (int i = 0; i < 16; ++i) {
            S[i] = 0.99f * S[i] - 0.01f * sk * kc[i] + kr * vc[i];
            o += S[i] * qc[i];
        }
        o += __shfl_xor(o, 1);
        o += __shfl_xor(o, 2);
        if (c == 0)
            yraw[(size_t)(b * TLEN + t) * CDIM + h * HS + r] = o;
        __syncthreads();
    }
}

// ---------------------------------------------------------------------------
extern "C" void kernel_launch(void* const* d_in, const int* in_sizes, int n_in,
                              void* d_out, int out_size, void* d_ws, size_t ws_size,
                              hipStream_t stream)
{
    const float* x      = (const float*)d_in[0];   // (4,1024,768)
    const float* w_attn = (const float*)d_in[1];   // (768, 2304)
    const float* w_proj = (const float*)d_in[2];   // (768, 768)
    float* out  = (float*)d_out;                   // (4,1024,768)

    const int M = BSZ * TLEN;                      // 4096
    float* qkv  = (float*)d_ws;                    // 4096 x 2304
    float* yraw = qkv + (size_t)M * (3 * CDIM);    // 4096 x 768

    // 1) qkv = x @ w_attn       64x64 tiles: (4096/64)*(2304/64) = 2304 blocks
    wmma_gemm_f32<<<2304, 128, 0, stream>>>(x, w_attn, nullptr, qkv,
                                            M, 3 * CDIM, CDIM);
    // 2) standardize each 64-element head-dim group in place
    const int nseg = M * (3 * CDIM) / 64;          // 147456
    qkv_norm<<<nseg / 8, 256, 0, stream>>>(qkv, nseg);
    // 3) sequential delta-rule scan, one block per (b,h)
    deltanet_scan<<<BSZ * NH, 256, 0, stream>>>(qkv, yraw);
    // 4) out = x + yraw @ w_proj    tiles: (4096/64)*(768/64) = 768 blocks
    wmma_gemm_f32<<<768, 128, 0, stream>>>(yraw, w_proj, x, out,
                                           M, CDIM, CDIM);
}